// LocalPatchEncoder_70171175682080
// MI455X (gfx1250) — compile-verified
//
#include <hip/hip_runtime.h>
#include <hip/hip_bf16.h>
#include <math.h>

#define B_ 8
#define N_ 16384
#define S_ 1024
#define K_ 32
#define CSTEM 64
#define CPATCH 128
#define INP 96            // 91 real input dims padded to 96 (multiple of 4 for K-steps)
#define R2 0.04f
#define PI_F 3.14159265358979323846f

typedef __attribute__((ext_vector_type(2))) float v2f;
typedef __attribute__((ext_vector_type(8))) float v8f;

// ---------------------------------------------------------------------------
// Kernel 1: ball query with early exit. One wave32 per (b,s) pair.
// Scans points in index order; ballot + prefix-popcount appends hits;
// stops once K found. Padding: remaining slots get first hit (or N-1).
// ---------------------------------------------------------------------------
__global__ __launch_bounds__(256) void knn_kernel(const float* __restrict__ xyz,
                                                  const float* __restrict__ center,
                                                  int* __restrict__ idx_out) {
    const int wave = (blockIdx.x * blockDim.x + threadIdx.x) >> 5;  // (b,s) pair
    const int lane = threadIdx.x & 31;
    const int b = wave >> 10;          // S_ = 1024
    const int s = wave & (S_ - 1);

    const float cx = center[((long)b * S_ + s) * 3 + 0];
    const float cy = center[((long)b * S_ + s) * 3 + 1];
    const float cz = center[((long)b * S_ + s) * 3 + 2];

    int* out = idx_out + (long)wave * K_;
    int count = 0;
    int first_idx = N_ - 1;            // covers the "no hit" case (min(gi, n-1))

    for (int base = 0; base < N_ && count < K_; base += 32) {
        const int i = base + lane;
        const float* p = xyz + ((long)b * N_ + i) * 3;
        const float dx = p[0] - cx, dy = p[1] - cy, dz = p[2] - cz;
        const bool cond = (dx * dx + dy * dy + dz * dz) <= R2;
        const unsigned long long m = __ballot(cond);   // wave32: low 32 bits
        if (count == 0 && m != 0ull)
            first_idx = base + (__ffsll(m) - 1);
        const int prefix = count + __popcll(m & ((1ull << lane) - 1ull));
        if (cond && prefix < K_) out[prefix] = i;
        count += (int)__popcll(m);
    }
    // pad remaining slots with first hit (or N-1 if none)
    for (int t = count + lane; t < K_; t += 32) out[t] = first_idx;
}

// ---------------------------------------------------------------------------
// Kernel 2: per-(b,s) MLP with fp32 WMMA (V_WMMA_F32_16X16X4_F32).
// 8 waves per block; wave w owns output columns [16w, 16w+16).
// ---------------------------------------------------------------------------
__global__ __launch_bounds__(256) void lpe_mlp(const float* __restrict__ xyz,
                                               const float* __restrict__ feat,
                                               const float* __restrict__ center,
                                               const float* __restrict__ w1,
                                               const float* __restrict__ b1,
                                               const float* __restrict__ w2,
                                               const float* __restrict__ b2,
                                               const int* __restrict__ nidx_all,
                                               float* __restrict__ out) {
    __shared__ __align__(16) float Xs[K_ * INP];      // 32 x 96 input tile (12 KB)
    __shared__ __align__(16) float H1[K_ * CPATCH];   // 32 x 128 hidden  (16 KB)
    __shared__ float relS[K_ * 3];
    __shared__ int   idxS[K_];

    const int bs   = blockIdx.x;       // 0..8191
    const int b    = bs >> 10;
    const int tid  = threadIdx.x;
    const int lane = tid & 31;
    const int wv   = tid >> 5;         // n-tile id 0..7
    const int cl   = lane & 15;        // column within 16-wide tile
    const int hi   = lane >> 4;        // lane half (selects K pair / M offset)
    const int n0   = wv * 16;

    // ---- Stage 0a: indices + relative xyz into LDS ----
    if (tid < K_) idxS[tid] = nidx_all[(long)bs * K_ + tid];
    __syncthreads();
    if (tid < K_ * 3) {
        const int row = tid / 3, d = tid - row * 3;
        relS[tid] = xyz[((long)b * N_ + idxS[row]) * 3 + d] - center[(long)bs * 3 + d];
    }
    __syncthreads();

    // ---- Stage 0b: build X = [feat(64) | rel(3) | PE(24) | 0-pad(5)] ----
    for (int e = tid; e < K_ * INP; e += 256) {
        const int row = e / INP, col = e - row * INP;
        float v;
        if (col < CSTEM) {
            v = feat[((long)b * N_ + idxS[row]) * CSTEM + col];
        } else if (col < 67) {
            v = relS[row * 3 + (col - 64)];
        } else if (col < 91) {
            const int t = col - 67;
            const int d = t >> 3, j = t & 7;          // per dim: sin f0..f3, cos f0..f3
            const float ang = relS[row * 3 + d] * ((float)(1 << (j & 3)) * PI_F);
            v = (j < 4) ? __sinf(ang) : __cosf(ang);
        } else {
            v = 0.0f;                                  // pad cols 91..95
        }
        Xs[e] = v;
    }
    __syncthreads();

    // ---- Stage 1: H1 = relu(X @ W1 + b1), per-wave 16-col tile, fp32 WMMA ----
    // A frag (16x4): lanes 0-15 hold K=4k+{0,1}, lanes 16-31 hold K=4k+{2,3}
    v8f acc0 = {};   // rows 0..15
    v8f acc1 = {};   // rows 16..31
    for (int k = 0; k < INP / 4; ++k) {
        const int kk = 4 * k + 2 * hi;
        const v2f a0 = *(const v2f*)&Xs[cl * INP + kk];
        const v2f a1 = *(const v2f*)&Xs[(16 + cl) * INP + kk];
        v2f bb;
        bb.x = (kk     < 91) ? w1[(long)kk       * CPATCH + n0 + cl] : 0.0f;
        bb.y = (kk + 1 < 91) ? w1[(long)(kk + 1) * CPATCH + n0 + cl] : 0.0f;
        acc0 = __builtin_amdgcn_wmma_f32_16x16x4_f32(false, a0, false, bb, (short)0, acc0, false, false);
        acc1 = __builtin_amdgcn_wmma_f32_16x16x4_f32(false, a1, false, bb, (short)0, acc1, false, false);
    }
    {
        const float bias1 = b1[n0 + cl];
        #pragma unroll
        for (int r = 0; r < 8; ++r) {
            const int row0 = r + 8 * hi;               // C/D layout: VGPR r -> M=r / r+8
            float h0 = acc0[r] + bias1; h0 = h0 > 0.0f ? h0 : 0.0f;
            float h1 = acc1[r] + bias1; h1 = h1 > 0.0f ? h1 : 0.0f;
            H1[row0 * CPATCH + n0 + cl]        = h0;
            H1[(16 + row0) * CPATCH + n0 + cl] = h1;
        }
    }
    __syncthreads();

    // ---- Stage 2: H2 = H1 @ W2 + b2 ; patch_feature = rowmax(H2) ----
    v8f d0 = {};
    v8f d1 = {};
    for (int k = 0; k < CPATCH / 4; ++k) {
        const int kk = 4 * k + 2 * hi;
        const v2f a0 = *(const v2f*)&H1[cl * CPATCH + kk];
        const v2f a1 = *(const v2f*)&H1[(16 + cl) * CPATCH + kk];
        v2f bb;
        bb.x = w2[(long)kk       * CPATCH + n0 + cl];
        bb.y = w2[(long)(kk + 1) * CPATCH + n0 + cl];
        d0 = __builtin_amdgcn_wmma_f32_16x16x4_f32(false, a0, false, bb, (short)0, d0, false, false);
        d1 = __builtin_amdgcn_wmma_f32_16x16x4_f32(false, a1, false, bb, (short)0, d1, false, false);
    }
    float m = -INFINITY;
    #pragma unroll
    for (int r = 0; r < 8; ++r) {
        m = fmaxf(m, d0[r]);
        m = fmaxf(m, d1[r]);
    }
    // combine the two lane halves (rows r vs r+8 of each M-tile live 16 lanes apart)
    m = fmaxf(m, __shfl_xor(m, 16, 32));
    if (hi == 0)
        out[(long)bs * CPATCH + n0 + cl] = m + b2[n0 + cl];
}

// ---------------------------------------------------------------------------
extern "C" void kernel_launch(void* const* d_in, const int* in_sizes, int n_in,
                              void* d_out, int out_size, void* d_ws, size_t ws_size,
                              hipStream_t stream) {
    const float* xyz    = (const float*)d_in[0];
    const float* feat   = (const float*)d_in[1];
    const float* center = (const float*)d_in[2];
    const float* w1     = (const float*)d_in[3];
    const float* b1     = (const float*)d_in[4];
    const float* w2     = (const float*)d_in[5];
    const float* b2     = (const float*)d_in[6];

    float* out_feat = (float*)d_out;                              // B*S*128 fp32
    int*   out_idx  = (int*)(out_feat + (size_t)B_ * S_ * CPATCH); // B*S*K int32

    dim3 blk(256);
    // 8192 waves, 8 waves per block
    knn_kernel<<<(B_ * S_ * 32) / 256, blk, 0, stream>>>(xyz, center, out_idx);
    lpe_mlp<<<B_ * S_, blk, 0, stream>>>(xyz, feat, center, w1, b1, w2, b2,
                                         out_idx, out_feat);
}